// SSDLoss_91139206021183
// MI455X (gfx1250) — compile-verified
//
#include <hip/hip_runtime.h>
#include <stdint.h>

#define Bsz 64
#define Dn  8732
#define Cc  81
#define AF  85                 // 4 loc + 81 conf
#define BD  (Bsz * Dn)         // 558848
#define NB1 Dn                 // kernel1 block count (64 anchors per block)
#define K1T 64
#define K2T 256

// ---------- CDNA5 async global->LDS helpers (gfx1250) ----------
__device__ __forceinline__ unsigned ldsOff(const void* p) {
  // generic pointer to LDS: low 32 bits are the LDS byte offset (aperture base low bits are 0)
  return (unsigned)(uintptr_t)p;
}
__device__ __forceinline__ void asyncLoadB128(unsigned lds, unsigned goff, const void* base) {
  asm volatile("global_load_async_to_lds_b128 %0, %1, %2"
               :: "v"(lds), "v"(goff), "s"(base) : "memory");
}
__device__ __forceinline__ void asyncWait0() {
  asm volatile("s_wait_asynccnt 0" ::: "memory");
}

// ---------- wave32 reductions ----------
__device__ __forceinline__ float waveSum(float v) {
#pragma unroll
  for (int o = 16; o > 0; o >>= 1) v += __shfl_xor(v, o, 32);
  return v;
}
__device__ __forceinline__ int waveSumI(int v) {
#pragma unroll
  for (int o = 16; o > 0; o >>= 1) v += __shfl_xor(v, o, 32);
  return v;
}

// float -> order-preserving uint key (ascending)
__device__ __forceinline__ unsigned fkey(float f) {
  unsigned u = __float_as_uint(f);
  return (u & 0x80000000u) ? ~u : (u | 0x80000000u);
}

// =================== kernel 1: streaming per-anchor math ===================
__global__ void k1(const int* __restrict__ posi,
                   const float* __restrict__ pred,
                   const float* __restrict__ tgt,
                   unsigned* __restrict__ keys, float* __restrict__ cearr,
                   float* __restrict__ blkLoc, float* __restrict__ blkCE)
{
  __shared__ __align__(16) float tp[K1T * AF];   // 21760 B
  __shared__ __align__(16) float tt[K1T * AF];   // 21760 B
  __shared__ float s1[2], s2[2];

  const int tid = threadIdx.x;
  const long long a0 = (long long)blockIdx.x * K1T;
  const float* pbase = pred + a0 * AF;           // byte offset mult of 16
  const float* tbase = tgt  + a0 * AF;
  const unsigned ltp = ldsOff(tp), ltt = ldsOff(tt);
  const int chunks = K1T * AF * 4 / 16;          // 1360 x 16B per array

  for (int i = tid; i < chunks; i += K1T)
    asyncLoadB128(ltp + i * 16, (unsigned)(i * 16), pbase);
  for (int i = tid; i < chunks; i += K1T)
    asyncLoadB128(ltt + i * 16, (unsigned)(i * 16), tbase);
  asyncWait0();
  __syncthreads();

  const float* p = &tp[tid * AF];   // stride 85 dwords: 21 mod 64 -> bank-conflict-free
  const float* q = &tt[tid * AF];

  // smooth L1 over 4 loc coords
  float sl1 = 0.f;
#pragma unroll
  for (int k = 0; k < 4; k++) {
    float d = p[k] - q[k];
    float ad = fabsf(d);
    sl1 += (ad < 1.f) ? 0.5f * d * d : (ad - 0.5f);
  }
  // log-sum-exp over 81 conf classes (accurate expf/logf to match jax log_softmax)
  float m = p[4];
  for (int k = 5; k < AF; k++) m = fmaxf(m, p[k]);
  float s = 0.f;
  for (int k = 4; k < AF; k++) s += expf(p[k] - m);
  float lse = m + logf(s);
  float bg = lse - p[AF - 1];                     // -logp[last class]
  // first-max argmax of target conf
  float best = q[4]; int lab = 0;
  for (int k = 1; k < Cc; k++) { float v = q[4 + k]; if (v > best) { best = v; lab = k; } }
  float ce = lse - p[4 + lab];

  long long a = a0 + tid;
  bool pos = posi[a] != 0;
  keys[a]  = pos ? 0u : fkey(bg);                 // pos => -inf => smallest key
  cearr[a] = ce;

  float ll = pos ? sl1 : 0.f;
  float cp = pos ? ce  : 0.f;
  ll = waveSum(ll); cp = waveSum(cp);
  int lane = tid & 31, wv = tid >> 5;
  if (lane == 0) { s1[wv] = ll; s2[wv] = cp; }
  __syncthreads();
  if (tid == 0) {
    blkLoc[blockIdx.x] = s1[0] + s1[1];
    blkCE[blockIdx.x]  = s2[0] + s2[1];
  }
}

// =================== kernel 2: per-row hard-negative selection ===================
__global__ void k2(const unsigned* __restrict__ keys,
                   const float* __restrict__ cearr,
                   float* __restrict__ rowNeg, int* __restrict__ rowPos)
{
  __shared__ __align__(16) unsigned sk[Dn];       // 34928 B
  __shared__ int   bins[256];
  __shared__ int   s_wcnt[8];
  __shared__ float s8[8];
  __shared__ unsigned s_pfx;
  __shared__ int   s_rem, s_taken;
  __shared__ float s_eqsum;

  const int tid = threadIdx.x;
  const int lane = tid & 31, wv = tid >> 5;
  const int b = blockIdx.x;
  const unsigned* gk = keys  + (size_t)b * Dn;    // byte offset mult of 16
  const float*    gc = cearr + (size_t)b * Dn;

  // async-stage row keys into LDS
  unsigned lsk = ldsOff(sk);
  const int chunks = Dn * 4 / 16;                 // 2183 x 16B
  for (int i = tid; i < chunks; i += K2T)
    asyncLoadB128(lsk + i * 16, (unsigned)(i * 16), gk);
  asyncWait0();
  __syncthreads();

  // positives: key == 0 (all negative keys >= 0x80000000 since bg >= 0)
  int pc = 0;
  for (int i = tid; i < Dn; i += K2T) pc += (sk[i] == 0u);
  pc = waveSumI(pc);
  if (lane == 0) s_wcnt[wv] = pc;
  __syncthreads();
  int pos = 0;
#pragma unroll
  for (int w = 0; w < 8; w++) pos += s_wcnt[w];

  int k = Dn - pos;
  int k3 = 3 * pos;
  if (k3 < k) k = k3;                             // neg_num
  if (tid == 0) rowPos[b] = pos;
  if (k <= 0) { if (tid == 0) rowNeg[b] = 0.f; return; }

  if (tid == 0) { s_pfx = 0u; s_rem = k; s_taken = 0; s_eqsum = 0.f; }
  __syncthreads();

  // MSB-first radix select of the k-th largest key
  for (int pass = 3; pass >= 0; --pass) {
    bins[tid] = 0;                                // blockDim == 256
    __syncthreads();
    unsigned pfx = s_pfx;
    int hiShift = (pass + 1) * 8;
    for (int i = tid; i < Dn; i += K2T) {
      unsigned key = sk[i];
      bool match = (pass == 3) || ((key >> hiShift) == (pfx >> hiShift));
      if (match) atomicAdd(&bins[(key >> (pass * 8)) & 0xFF], 1);
    }
    __syncthreads();
    if (tid == 0) {
      int rem = s_rem, cum = 0;
      for (int v = 255; v >= 0; --v) {
        cum += bins[v];
        if (cum >= rem) {
          s_pfx |= (unsigned)v << (pass * 8);
          s_rem = rem - (cum - bins[v]);
          break;
        }
      }
    }
    __syncthreads();
  }
  const unsigned T = s_pfx;
  const int quota = s_rem;                        // ties at T to take, in index order

  // sum ce where key > T
  float loc = 0.f;
  for (int i = tid; i < Dn; i += K2T)
    if (sk[i] > T) loc += gc[i];
  loc = waveSum(loc);
  if (lane == 0) s8[wv] = loc;
  __syncthreads();
  float sum_gt = 0.f;
  if (tid == 0) { for (int w = 0; w < 8; w++) sum_gt += s8[w]; }
  __syncthreads();

  // stable (index-ordered) selection among key == T, matching jnp stable argsort
  for (int base = 0; base < Dn; base += K2T) {
    int cur = s_taken;                            // uniform (after prior barrier)
    if (cur >= quota) break;                      // uniform break
    int i = base + tid;
    bool eq = (i < Dn) && (sk[i] == T);
    unsigned msk = (unsigned)__ballot(eq);
    if (lane == 0) s_wcnt[wv] = __popc(msk);
    __syncthreads();
    int woff = 0;
    for (int w = 0; w < wv; w++) woff += s_wcnt[w];
    int ctot = 0;
#pragma unroll
    for (int w = 0; w < 8; w++) ctot += s_wcnt[w];
    int rank = cur + woff + __popc(msk & ((1u << lane) - 1u));
    float add = (eq && rank < quota) ? gc[i] : 0.f;
    add = waveSum(add);
    if (lane == 0) s8[wv] = add;
    __syncthreads();
    if (tid == 0) {
      float r = 0.f;
      for (int w = 0; w < 8; w++) r += s8[w];
      s_eqsum += r;
      s_taken = cur + ctot;
    }
    __syncthreads();
  }
  if (tid == 0) rowNeg[b] = sum_gt + s_eqsum;
}

// =================== kernel 3: final deterministic combine ===================
__global__ void k3(const float* __restrict__ blkLoc, const float* __restrict__ blkCE,
                   const float* __restrict__ rowNeg, const int* __restrict__ rowPos,
                   float* __restrict__ out)
{
  __shared__ float s8[8];
  __shared__ int   si8[8];
  const int tid = threadIdx.x, lane = tid & 31, wv = tid >> 5;
  float accL = 0.f, accC = 0.f, accN = 0.f; int accP = 0;
  for (int i = tid; i < NB1; i += 256) { accL += blkLoc[i]; accC += blkCE[i]; }
  if (tid < Bsz) { accN = rowNeg[tid]; accP = rowPos[tid]; }
  accL = waveSum(accL); accC = waveSum(accC); accN = waveSum(accN); accP = waveSumI(accP);

  if (lane == 0) s8[wv] = accL;
  __syncthreads();
  float L = 0.f; if (tid == 0) for (int w = 0; w < 8; w++) L += s8[w];
  __syncthreads();
  if (lane == 0) s8[wv] = accC;
  __syncthreads();
  float Cp = 0.f; if (tid == 0) for (int w = 0; w < 8; w++) Cp += s8[w];
  __syncthreads();
  if (lane == 0) s8[wv] = accN;
  __syncthreads();
  float Ng = 0.f; if (tid == 0) for (int w = 0; w < 8; w++) Ng += s8[w];
  __syncthreads();
  if (lane == 0) si8[wv] = accP;
  __syncthreads();
  if (tid == 0) {
    int P = 0;
    for (int w = 0; w < 8; w++) P += si8[w];
    float N = (float)P;
    out[0] = (Cp + Ng) / N;   // conf_loss
    out[1] = L / N;           // loc_loss
  }
}

// =================== launcher ===================
extern "C" void kernel_launch(void* const* d_in, const int* in_sizes, int n_in,
                              void* d_out, int out_size, void* d_ws, size_t ws_size,
                              hipStream_t stream)
{
  const int*   posi = (const int*)d_in[0];     // pos_indicator (bool -> int)
  const float* pred = (const float*)d_in[1];   // (B, D, 85)
  const float* tgt  = (const float*)d_in[2];   // (B, D, 85)
  float* out = (float*)d_out;

  uint8_t* ws = (uint8_t*)d_ws;
  unsigned* keys  = (unsigned*)ws;                                        // BD u32
  float*   ce     = (float*)(ws + (size_t)4 * BD);                        // BD f32
  float*   blkLoc = (float*)(ws + (size_t)8 * BD);                        // NB1 f32
  float*   blkCE  = (float*)(ws + (size_t)8 * BD + 4 * NB1);              // NB1 f32
  float*   rowNeg = (float*)(ws + (size_t)8 * BD + 8 * NB1);              // Bsz f32
  int*     rowPos = (int*)  (ws + (size_t)8 * BD + 8 * NB1 + 4 * Bsz);    // Bsz i32

  k1<<<NB1, K1T, 0, stream>>>(posi, pred, tgt, keys, ce, blkLoc, blkCE);
  k2<<<Bsz, K2T, 0, stream>>>(keys, ce, rowNeg, rowPos);
  k3<<<1, 256, 0, stream>>>(blkLoc, blkCE, rowNeg, rowPos, out);
}